// StatisticsPooling_41343355191551
// MI455X (gfx1250) — compile-verified
//
#include <hip/hip_runtime.h>
#include <hip/hip_bf16.h>
#include <cstdint>
#include <cstddef>

// Problem constants (match the reference).
#define B_DIM 16
#define T_DIM 2048
#define F_DIM 768
#define EPSV  1e-5f

typedef __attribute__((ext_vector_type(2))) float v2f;
typedef __attribute__((ext_vector_type(8))) float v8f;

// Deterministic hash -> uniform [0,1). Stand-in for the reference's
// (unreplicable-without-runtime) JAX threefry gaussian-noise term; same
// magnitude structure: noise = EPS * (9 - 8*g), g in [0,1].
__device__ __forceinline__ float hash_unit(uint32_t x) {
    x ^= x >> 16; x *= 0x7feb352dU;
    x ^= x >> 15; x *= 0x846ca68bU;
    x ^= x >> 16;
    return (float)(x >> 8) * (1.0f / 16777216.0f);
}

// ---------------------------------------------------------------------------
// Kernel 1: per-batch valid-prefix length.
//   seq_len[b] = sum_t mask[b,t];  actual[b] = rint( (len/max_len) * T )
// One block, 16 waves (wave32) -> one wave per batch row.
// ---------------------------------------------------------------------------
__global__ void __launch_bounds__(512)
seqlen_kernel(const int* __restrict__ mask, int* __restrict__ actual) {
    __shared__ int lens[B_DIM];
    const int wave = threadIdx.x >> 5;   // 0..15 == batch index
    const int lane = threadIdx.x & 31;

    const int* row = mask + (size_t)wave * T_DIM;
    int s = 0;
    for (int t = lane; t < T_DIM; t += 32) s += row[t];
#pragma unroll
    for (int off = 16; off >= 1; off >>= 1) s += __shfl_xor(s, off, 32);
    if (lane == 0) lens[wave] = s;
    __syncthreads();

    if (threadIdx.x == 0) {
        int mx = 1;
#pragma unroll
        for (int b = 0; b < B_DIM; ++b) mx = (lens[b] > mx) ? lens[b] : mx;
#pragma unroll
        for (int b = 0; b < B_DIM; ++b) {
            const float rel = (float)lens[b] / (float)mx;
            actual[b] = (int)rintf(rel * (float)T_DIM);  // half-to-even, like jnp.round
        }
    }
}

// ---------------------------------------------------------------------------
// Kernel 2: masked sum / sum-of-squares over T via V_WMMA_F32_16X16X4_F32.
//
// Per block: one (b, 16-wide f-tile). 8 waves, each owning a 256-token slice.
// WMMA operands per K=4 step:
//   A (16x4, f32): row M=0 carries the prefix mask m[t..t+3]; rows 1..15 zero.
//     Layout: lanes 0-15 = M, VGPR0=K0 VGPR1=K1; lanes 16-31 = M, K2/K3
//     => only lane 0 (K0,K1) and lane 16 (K2,K3) carry nonzero A values.
//   B (4x16, f32): x[t+k, f0+n] tile (second WMMA uses x^2 elementwise).
//     Mirror of A: lanes 0-15 hold (K0,K1) for N=lane, lanes 16-31 hold
//     (K2,K3) for N=lane-16.
// Accumulator row M=0 (VGPR0, lanes 0-15) = per-f partial sum over the slice.
//
// The token slice is split into a fully-valid region (mask hoisted to a
// constant A fragment -> no per-iteration compares) and a masked tail
// (<=1 straddling K-group; waves fully past `actual` break immediately,
// skipping their loads entirely). All branches are wave-uniform, so EXEC
// is all-ones around every WMMA.
// ---------------------------------------------------------------------------
__global__ void __launch_bounds__(256)
stats_pool_kernel(const float* __restrict__ x,
                  const int*  __restrict__ actual_arr,
                  float* __restrict__ out) {
    const int f0   = blockIdx.x * 16;
    const int b    = blockIdx.y;
    const int wave = threadIdx.x >> 5;
    const int lane = threadIdx.x & 31;
    const int half = lane >> 4;     // 0 -> (K0,K1) rows, 1 -> (K2,K3) rows
    const int l16  = lane & 15;

    const int actual = actual_arr[b];
    const float* __restrict__ xb = x + (size_t)b * T_DIM * F_DIM;

    v8f csum = {};
    v8f csq  = {};

    const int tspan = T_DIM / 8;        // 256 tokens per wave
    const int tbeg  = wave * tspan;
    const int tend  = tbeg + tspan;

    // K-groups [t, t+4) with t+4 <= actual are fully valid.
    int full_end = (actual >> 2) << 2;
    if (full_end > tend) full_end = tend;
    if (full_end < tbeg) full_end = tbeg;

    // Hoisted constant A fragment for the fully-valid region (row M=0 ones).
    v2f a1;
    a1.x = (l16 == 0) ? 1.0f : 0.0f;
    a1.y = a1.x;

    int t = tbeg;
    for (; t < full_end; t += 4) {
        const int tk = t + (half << 1);
        v2f bv;
        bv.x = xb[(size_t)(tk    ) * F_DIM + f0 + l16];
        bv.y = xb[(size_t)(tk + 1) * F_DIM + f0 + l16];
        v2f b2 = bv * bv;
        csum = __builtin_amdgcn_wmma_f32_16x16x4_f32(
            false, a1, false, bv, (short)0, csum, false, false);
        csq  = __builtin_amdgcn_wmma_f32_16x16x4_f32(
            false, a1, false, b2, (short)0, csq,  false, false);
    }

    // Masked tail: at most one straddling K-group; uniform break otherwise.
    for (; t < tend; t += 4) {
        if (t >= actual) break;          // wave-uniform exit
        const int tk = t + (half << 1);
        v2f am;
        am.x = (l16 == 0 && (tk    ) < actual) ? 1.0f : 0.0f;
        am.y = (l16 == 0 && (tk + 1) < actual) ? 1.0f : 0.0f;
        v2f bv;
        bv.x = xb[(size_t)(tk    ) * F_DIM + f0 + l16];
        bv.y = xb[(size_t)(tk + 1) * F_DIM + f0 + l16];
        v2f b2 = bv * bv;
        csum = __builtin_amdgcn_wmma_f32_16x16x4_f32(
            false, am, false, bv, (short)0, csum, false, false);
        csq  = __builtin_amdgcn_wmma_f32_16x16x4_f32(
            false, am, false, b2, (short)0, csq,  false, false);
    }

    // Cross-wave combine of row M=0 (accumulator VGPR0, lanes 0-15).
    __shared__ float red[2][8][16];
    if (lane < 16) {
        red[0][wave][lane] = csum[0];
        red[1][wave][lane] = csq[0];
    }
    __syncthreads();

    if (threadIdx.x < 16) {
        float s1 = 0.0f, s2 = 0.0f;
#pragma unroll
        for (int w = 0; w < 8; ++w) {
            s1 += red[0][w][threadIdx.x];
            s2 += red[1][w][threadIdx.x];
        }
        const float cnt  = fmaxf((float)actual, 1.0f);
        const float mean = s1 / cnt;
        float var = (s2 - cnt * mean * mean) / fmaxf(cnt - 1.0f, 1.0f);
        var = fmaxf(var, 0.0f);

        const int f = f0 + threadIdx.x;
        const float g = hash_unit((uint32_t)(b * F_DIM + f) * 2654435761u + 12345u);

        float* __restrict__ ob = out + (size_t)b * (2 * F_DIM);
        ob[f]         = mean + EPSV * (9.0f - 8.0f * g);  // mean + gauss-noise term
        ob[F_DIM + f] = sqrtf(var) + EPSV;                // std + EPS
    }
}

// ---------------------------------------------------------------------------
// Host-side launcher.
//   d_in[0]: input_values  f32 [16, 2048, 768]
//   d_in[1]: attention_mask i32 [16, 2048]
//   d_out  : f32 [16, 1, 1536] (mean || std)
//   d_ws   : scratch (16 ints for actual[])
// ---------------------------------------------------------------------------
extern "C" void kernel_launch(void* const* d_in, const int* in_sizes, int n_in,
                              void* d_out, int out_size, void* d_ws, size_t ws_size,
                              hipStream_t stream) {
    (void)in_sizes; (void)n_in; (void)out_size; (void)ws_size;

    const float* x      = (const float*)d_in[0];
    const int*   mask   = (const int*)d_in[1];
    float*       out    = (float*)d_out;
    int*         actual = (int*)d_ws;

    seqlen_kernel<<<1, 512, 0, stream>>>(mask, actual);

    dim3 grid(F_DIM / 16, B_DIM);   // 48 f-tiles x 16 batches
    stats_pool_kernel<<<grid, 256, 0, stream>>>(x, actual, out);
}